// SemanticSubcharacterAugmentation_30674656428480
// MI455X (gfx1250) — compile-verified
//
#include <hip/hip_runtime.h>

// SemanticSubcharacterAugmentation for MI455X (gfx1250, wave32).
//
// out[b,e,0,l] = x[b,e,0,l] + (e == idx[b,l] ? (2*u[b,e,0,l]-1)*GAMMA : 0)
//
// Scatter indices are collision-free, so this is a streaming copy with a
// sparse fused add. HBM-bound: ~256 MiB of mandatory traffic -> ~11.5 us at
// 23.3 TB/s. Strategy: b128 NT loads/stores for the streaming x/out data,
// cached (RT) 128-bit idx loads (re-used 256x across e, L2-resident),
// exec-masked scalar u loads only on the 1/256 matching lanes.

typedef float vf4 __attribute__((ext_vector_type(4)));
typedef int   vi4 __attribute__((ext_vector_type(4)));

namespace {
constexpr int       kB = 128;
constexpr int       kE = 256;
constexpr int       kL = 1024;
constexpr float     kGamma = 0.2f;
constexpr long long kN = (long long)kB * kE * kL;   // 33,554,432 elements
}  // namespace

__global__ __launch_bounds__(256) void ssa_fused_kernel(
    const float* __restrict__ x,
    const float* __restrict__ u,
    const int*   __restrict__ idx,
    float*       __restrict__ out) {
  const long long t = (long long)blockIdx.x * blockDim.x + threadIdx.x;
  const long long g = t << 2;                 // flat element index, 16B aligned
  if (g >= kN) return;

  // Flat layout of (B, E, 1, L): offset = (b*E + e)*L + l
  const int       l = (int)(g & (kL - 1));    // L = 1024
  const long long r = g >> 10;                // row = b*E + e
  const int       e = (int)(r & (kE - 1));    // E = 256
  const long long b = r >> 8;

  // idx[b, l..l+3]: 16B-aligned, regular temporal load (hot in L0/L2 —
  // re-read by all 256 e-slices of this (b,l) tile).
  const vi4 iv = *(const vi4*)(idx + b * (long long)kL + l);

  // Streaming 128-bit non-temporal load of x.
  vf4 xv = __builtin_nontemporal_load((const vf4*)(x + g));

  // Sparse fused perturbation: (2u-1)*gamma = fma(u, 2*gamma, -gamma).
  // Exec-masked scalar loads of u fire only on matching lanes (~1/256).
  if (iv.x == e) xv.x += __builtin_fmaf(u[g + 0], 2.0f * kGamma, -kGamma);
  if (iv.y == e) xv.y += __builtin_fmaf(u[g + 1], 2.0f * kGamma, -kGamma);
  if (iv.z == e) xv.z += __builtin_fmaf(u[g + 2], 2.0f * kGamma, -kGamma);
  if (iv.w == e) xv.w += __builtin_fmaf(u[g + 3], 2.0f * kGamma, -kGamma);

  // Streaming 128-bit non-temporal store.
  __builtin_nontemporal_store(xv, (vf4*)(out + g));
}

extern "C" void kernel_launch(void* const* d_in, const int* in_sizes, int n_in,
                              void* d_out, int out_size, void* d_ws, size_t ws_size,
                              hipStream_t stream) {
  const float* x   = (const float*)d_in[0];   // (B, E, 1, L) f32
  const float* u   = (const float*)d_in[1];   // (B, E, 1, L) f32
  const int*   idx = (const int*)d_in[2];     // (B, L) int32
  float*       out = (float*)d_out;           // (B, E, 1, L) f32

  const long long quads  = kN >> 2;           // one float4 per thread
  const int       block  = 256;               // 8 wave32 waves / block
  const long long nblk   = (quads + block - 1) / block;   // 32768

  ssa_fused_kernel<<<dim3((unsigned)nblk), dim3(block), 0, stream>>>(x, u, idx, out);
}